// MomentVideo_46943992545521
// MI455X (gfx1250) — compile-verified
//
#include <hip/hip_runtime.h>
#include <hip/hip_bf16.h>

typedef float v2f __attribute__((ext_vector_type(2)));
typedef float v4f __attribute__((ext_vector_type(4)));
typedef float v8f __attribute__((ext_vector_type(8)));

// D = A(16x4) * B(4x16) + C, fp32 WMMA (CDNA5 V_WMMA_F32_16X16X4_F32)
__device__ __forceinline__ v8f wmma_f32(v2f a, v2f b, v8f c) {
    return __builtin_amdgcn_wmma_f32_16x16x4_f32(
        /*neg_a=*/false, a, /*neg_b=*/false, b,
        /*c_mod=*/(short)0, c, /*reuse_a=*/false, /*reuse_b=*/false);
}

// ---------------------------------------------------------------------------
// Kernel 1: C[M,N] = A[M,K] @ W[K,N] + bias[N], row-major.
// Requires M%128==0, N%64==0, K%32==0.
// Block 256 threads (8 waves), 128x64 C tile, K chunk 32.
// Wave w: 32x32 sub-tile at (wm=(w&3)*32, wn=(w>>2)*32) = 4 WMMA tiles.
// B is staged TRANSPOSED in LDS so both A and B fragments are contiguous
// 8-byte reads (ds_load_b64), no v_mov reassembly.
// Per k-step: 2 A-frags + 2 B-frags feed 4 WMMAs (frag:wmma ratio 1.0).
// ---------------------------------------------------------------------------
#define BM 128
#define BN 64
#define BK 32
#define AS_STRIDE 36   // even (b64 align), 36n mod 64 distinct for 16 lanes
#define BS_STRIDE 34   // even (b64 align), 34n mod 64 distinct for 16 lanes

__global__ __launch_bounds__(256) void gemm_nn_bias(
    const float* __restrict__ A, const float* __restrict__ Bw,
    const float* __restrict__ bias, float* __restrict__ C,
    int M, int N, int K)
{
    __shared__ float As[BM][AS_STRIDE];   // As[m][k]
    __shared__ float BsT[BN][BS_STRIDE];  // BsT[n][k]  (transposed)

    const int tid  = threadIdx.x;
    const int wave = tid >> 5;
    const int lane = tid & 31;
    const int nblk_n = N / BN;
    const int row0 = (blockIdx.x / nblk_n) * BM;
    const int col0 = (blockIdx.x % nblk_n) * BN;
    const int wm = (wave & 3) * 32;
    const int wn = (wave >> 2) * 32;
    const int lhalf = lane & 15;
    const int khalf = (lane >> 4) * 2;   // 0 or 2

    v8f acc00 = {}, acc01 = {}, acc10 = {}, acc11 = {};

    for (int k0 = 0; k0 < K; k0 += BK) {
        // A tile: 128x32 floats = 1024 float4, 4 per thread (b128, coalesced)
        #pragma unroll
        for (int i = 0; i < 4; ++i) {
            int idx = tid + 256 * i;
            int r = idx >> 3, c4 = (idx & 7) * 4;
            *(v4f*)&As[r][c4] = *(const v4f*)&A[(size_t)(row0 + r) * K + k0 + c4];
        }
        // B tile: 32x64 floats = 512 float4, 2 per thread; store transposed
        #pragma unroll
        for (int i = 0; i < 2; ++i) {
            int idx = tid + 256 * i;
            int r = idx >> 4, c4 = (idx & 15) * 4;
            v4f v = *(const v4f*)&Bw[(size_t)(k0 + r) * N + col0 + c4];
            BsT[c4 + 0][r] = v.x;
            BsT[c4 + 1][r] = v.y;
            BsT[c4 + 2][r] = v.z;
            BsT[c4 + 3][r] = v.w;
        }
        __syncthreads();
        #pragma unroll
        for (int kk = 0; kk < BK; kk += 4) {
            // lanes 0-15 -> (k,k+1); lanes 16-31 -> (k+2,k+3)
            v2f a0 = *(const v2f*)&As [wm + lhalf]     [kk + khalf];
            v2f a1 = *(const v2f*)&As [wm + 16 + lhalf][kk + khalf];
            v2f b0 = *(const v2f*)&BsT[wn + lhalf]     [kk + khalf];
            v2f b1 = *(const v2f*)&BsT[wn + 16 + lhalf][kk + khalf];
            acc00 = wmma_f32(a0, b0, acc00);
            acc01 = wmma_f32(a0, b1, acc01);
            acc10 = wmma_f32(a1, b0, acc10);
            acc11 = wmma_f32(a1, b1, acc11);
        }
        __syncthreads();
    }

    const float bv0 = bias[col0 + wn + lhalf];
    const float bv1 = bias[col0 + wn + 16 + lhalf];
    const int r0 = row0 + wm + (lane >> 4) * 8;        // C: lanes 16-31 rows +8
    const int r1 = r0 + 16;
    const size_t cb0 = (size_t)col0 + wn + lhalf;
    const size_t cb1 = cb0 + 16;
    #pragma unroll
    for (int r = 0; r < 8; ++r) {
        C[(size_t)(r0 + r) * N + cb0] = acc00[r] + bv0;
        C[(size_t)(r0 + r) * N + cb1] = acc01[r] + bv1;
        C[(size_t)(r1 + r) * N + cb0] = acc10[r] + bv0;
        C[(size_t)(r1 + r) * N + cb1] = acc11[r] + bv1;
    }
}

// ---------------------------------------------------------------------------
// Kernel 2: scores[b,l,t] = sum_d Sr[b,l,d] * Vr[b,t,d]   (NT GEMM, batched)
// grid = B*25, block 64 (2 waves). Wave w: rows 16w..16w+15, cols t0..t0+15.
// Both fragments are contiguous v2f LDS reads (B staged as Vs[n][k]).
// ---------------------------------------------------------------------------
__global__ __launch_bounds__(64) void scores_nt(
    const float* __restrict__ Sr, const float* __restrict__ Vr,
    float* __restrict__ Sc)
{
    __shared__ float Ss[32][36];
    __shared__ float Vs[16][36];   // Vs[n][k] = Vr[t0+n][k0+k]

    const int b  = blockIdx.x / 25;
    const int t0 = (blockIdx.x % 25) * 16;
    const float* Srb = Sr + (size_t)b * 32 * 1024;
    const float* Vrb = Vr + (size_t)b * 400 * 1024;
    const int tid = threadIdx.x, wave = tid >> 5, lane = tid & 31;
    const int lhalf = lane & 15, khalf = (lane >> 4) * 2;

    v8f acc = {};
    for (int k0 = 0; k0 < 1024; k0 += 32) {
        #pragma unroll
        for (int i = 0; i < 4; ++i) {          // Ss: 32x32 = 256 float4
            int idx = tid + 64 * i;
            int r = idx >> 3, c4 = (idx & 7) * 4;
            *(v4f*)&Ss[r][c4] = *(const v4f*)&Srb[(size_t)r * 1024 + k0 + c4];
        }
        #pragma unroll
        for (int i = 0; i < 2; ++i) {          // Vs: 16x32 = 128 float4
            int idx = tid + 64 * i;
            int r = idx >> 3, c4 = (idx & 7) * 4;
            *(v4f*)&Vs[r][c4] = *(const v4f*)&Vrb[(size_t)(t0 + r) * 1024 + k0 + c4];
        }
        __syncthreads();
        #pragma unroll
        for (int kk = 0; kk < 32; kk += 4) {
            v2f a  = *(const v2f*)&Ss[wave * 16 + lhalf][kk + khalf];
            v2f bb = *(const v2f*)&Vs[lhalf][kk + khalf];   // B[k][n]=Vs[n][k]
            acc = wmma_f32(a, bb, acc);
        }
        __syncthreads();
    }
    const int rbase = wave * 16 + (lane >> 4) * 8;
    #pragma unroll
    for (int r = 0; r < 8; ++r)
        Sc[((size_t)b * 32 + rbase + r) * 400 + t0 + lhalf] = acc[r];
}

// ---------------------------------------------------------------------------
// Kernel 3: one block per (b, proposal). Softmax window -> LDS, then
// out = attn(32 x W) @ Vr[window](W x 1024) via WMMA; cosine-sim reductions
// straight out of the accumulators. positive for matched proposal, per-(b,p)
// max of score elsewhere. Vr re-reads are L2-resident (26 MB << 192 MB).
// ---------------------------------------------------------------------------
#define ATT_STRIDE 404   // 404 % 64 = 20 -> conflict-free 16-lane row reads

__global__ __launch_bounds__(256) void proposal_attend(
    const float* __restrict__ Sc,     // [16][32][400]
    const float* __restrict__ Vr,     // [16][400][1024]
    const float* __restrict__ Sr,     // [16][32][1024]
    const int*   __restrict__ y_true, // [16][2]
    float* __restrict__ positive,     // [16][1024]
    float* __restrict__ pmax)         // [16][136]
{
    __shared__ float att[32 * ATT_STRIDE];
    __shared__ float red[256];

    const int b = blockIdx.x / 136;
    const int p = blockIdx.x % 136;
    // decode proposal p -> (s,e); proposals enumerated s ascending, e=s..15
    int pp = p, s = 0;
    while (pp >= 16 - s) { pp -= 16 - s; ++s; }
    const int e  = s + pp;
    const int ts = s * 25, te = (e + 1) * 25;
    const int W  = te - ts;
    const int Wpad = (W + 3) & ~3;
    // ground-truth match index
    const int ys = y_true[2 * b], ye = y_true[2 * b + 1];
    const int mi = ys * 16 - (ys * (ys - 1)) / 2 + (ye - ys);
    const bool is_match = (mi == p);

    const int tid = threadIdx.x, wave = tid >> 5, lane = tid & 31;

    // ---- softmax over window, 4 rows per wave, lanes stride over t ----
    const float* Scb = Sc + (size_t)b * 32 * 400;
    for (int li = 0; li < 4; ++li) {
        const int l = wave * 4 + li;
        const float* row = Scb + (size_t)l * 400 + ts;
        float m = -__builtin_inff();
        for (int t = lane; t < W; t += 32) m = fmaxf(m, row[t]);
        for (int off = 16; off >= 1; off >>= 1) m = fmaxf(m, __shfl_xor(m, off, 32));
        float ssum = 0.f;
        for (int t = lane; t < W; t += 32) ssum += __expf(row[t] - m);
        for (int off = 16; off >= 1; off >>= 1) ssum += __shfl_xor(ssum, off, 32);
        const float inv = 1.0f / ssum;
        for (int t = lane; t < Wpad; t += 32)
            att[l * ATT_STRIDE + t] = (t < W) ? __expf(row[t] - m) * inv : 0.f;
    }
    __syncthreads();

    const float* Vrb = Vr + (size_t)b * 400 * 1024;
    const float* Srb = Sr + (size_t)b * 32 * 1024;
    const int lhalf = lane & 15;
    const int khalf = (lane >> 4) * 2;
    float negmax = -__builtin_inff();

    // 1024 cols in 8 chunks of 128; wave owns 16 cols; M=32 -> 2 WMMA tiles
    for (int c = 0; c < 8; ++c) {
        const int col = c * 128 + wave * 16 + lhalf;
        v8f acc0 = {}; v8f acc1 = {};
        for (int t = 0; t < Wpad; t += 4) {
            v2f a0 = *(const v2f*)&att[(lhalf)      * ATT_STRIDE + t + khalf];
            v2f a1 = *(const v2f*)&att[(16 + lhalf) * ATT_STRIDE + t + khalf];
            // B frag: rows ts+t+khalf, ts+t+khalf+1 (clamped; attn=0 past W)
            const int r0 = min(ts + t + khalf,     399);
            const int r1 = min(ts + t + khalf + 1, 399);
            v2f bb;
            bb.x = Vrb[(size_t)r0 * 1024 + col];
            bb.y = Vrb[(size_t)r1 * 1024 + col];
            acc0 = wmma_f32(a0, bb, acc0);
            acc1 = wmma_f32(a1, bb, acc1);
        }
        // cosine-sim pieces for this lane's column (lane c & c+16 same col)
        const int rbase = (lane >> 4) * 8;
        float num = 0.f, sq = 0.f, srq = 0.f;
        #pragma unroll
        for (int r = 0; r < 8; ++r) {
            const float s0 = Srb[(size_t)(rbase + r) * 1024 + col];
            const float s1 = Srb[(size_t)(16 + rbase + r) * 1024 + col];
            num += acc0[r] * s0 + acc1[r] * s1;
            sq  += acc0[r] * acc0[r] + acc1[r] * acc1[r];
            srq += s0 * s0 + s1 * s1;
        }
        num += __shfl_xor(num, 16, 32);
        sq  += __shfl_xor(sq,  16, 32);
        srq += __shfl_xor(srq, 16, 32);
        const float score = num / (sqrtf(sq) * sqrtf(srq) + 1e-15f);
        if (is_match) {
            if (lane < 16) positive[(size_t)b * 1024 + col] = score;
        } else {
            negmax = fmaxf(negmax, score);
        }
    }

    red[tid] = negmax;
    __syncthreads();
    for (int st = 128; st > 0; st >>= 1) {
        if (tid < st) red[tid] = fmaxf(red[tid], red[tid + st]);
        __syncthreads();
    }
    if (tid == 0) pmax[b * 136 + p] = is_match ? -__builtin_inff() : red[0];
}

// ---------------------------------------------------------------------------
// Kernel 4: negative[b] = max over proposals of pmax[b][p]
// ---------------------------------------------------------------------------
__global__ __launch_bounds__(256) void reduce_neg(
    const float* __restrict__ pmax, float* __restrict__ out_neg)
{
    __shared__ float red[256];
    const int b = blockIdx.x, tid = threadIdx.x;
    red[tid] = (tid < 136) ? pmax[b * 136 + tid] : -__builtin_inff();
    __syncthreads();
    for (int st = 128; st > 0; st >>= 1) {
        if (tid < st) red[tid] = fmaxf(red[tid], red[tid + st]);
        __syncthreads();
    }
    if (tid == 0) out_neg[b] = red[0];
}

// ---------------------------------------------------------------------------
extern "C" void kernel_launch(void* const* d_in, const int* in_sizes, int n_in,
                              void* d_out, int out_size, void* d_ws, size_t ws_size,
                              hipStream_t stream) {
    const float* videos    = (const float*)d_in[0];  // [16,400,4096]
    const float* sentences = (const float*)d_in[1];  // [16,32,768]
    const float* Wv        = (const float*)d_in[2];  // [4096,1024]
    const float* bv        = (const float*)d_in[3];  // [1024]
    const float* Ws        = (const float*)d_in[4];  // [768,1024]
    const float* bs        = (const float*)d_in[5];  // [1024]
    const int*   y_true    = (const int*)d_in[6];    // [16,2]
    float* out = (float*)d_out;                      // [16*1024] positive + [16] negative

    float* Vr   = (float*)d_ws;                      // 6400*1024
    float* Sr   = Vr + (size_t)6400 * 1024;          // 512*1024
    float* Sc   = Sr + (size_t)512 * 1024;           // 16*32*400
    float* pmax = Sc + (size_t)16 * 32 * 400;        // 16*136

    // Vr = videos @ Wv + bv : M=6400, N=1024, K=4096
    gemm_nn_bias<<<dim3((6400 / BM) * (1024 / BN)), dim3(256), 0, stream>>>(
        videos, Wv, bv, Vr, 6400, 1024, 4096);
    // Sr = sentences @ Ws + bs : M=512, N=1024, K=768
    gemm_nn_bias<<<dim3((512 / BM) * (1024 / BN)), dim3(256), 0, stream>>>(
        sentences, Ws, bs, Sr, 512, 1024, 768);
    // scores = Sr @ Vr^T per batch
    scores_nt<<<dim3(16 * 25), dim3(64), 0, stream>>>(Sr, Vr, Sc);
    // per-proposal attention + cosine similarity + positive/negmax
    proposal_attend<<<dim3(16 * 136), dim3(256), 0, stream>>>(
        Sc, Vr, Sr, y_true, out, pmax);
    // negative[b]
    reduce_neg<<<dim3(16), dim3(256), 0, stream>>>(pmax, out + 16 * 1024);
}